// RGATModel_2628519985724
// MI455X (gfx1250) — compile-verified
//
#include <hip/hip_runtime.h>

// Problem constants (match reference)
#define NNODES 50000
#define NEDGES 800000
#define DIN    128
#define CHID   64
#define NH     4
#define NREL   64
#define HC     256          // NH * 64, same both layers
#define NEG_SLOPE 0.2f

typedef __attribute__((ext_vector_type(2))) float v2f;
typedef __attribute__((ext_vector_type(8))) float v8f;

// ---- monotonic float <-> uint mapping for lock-free float atomicMax ----
__device__ __forceinline__ unsigned f32_ord(float f) {
  unsigned u = __float_as_uint(f);
  return (u & 0x80000000u) ? ~u : (u | 0x80000000u);
}
__device__ __forceinline__ float ord_f32(unsigned u) {
  return (u & 0x80000000u) ? __uint_as_float(u ^ 0x80000000u)
                           : __uint_as_float(~u);
}

// =====================================================================
// GEMM  C[M x 256] = A[M x din] * B[din x 256] using V_WMMA_F32_16X16X4_F32
// grid: (M/16, 2), block: 256 (8 waves). Wave w covers cols (by*8+w)*16.
// A panel (16 x din) staged once in LDS (ReLU optionally fused on load).
// =====================================================================
__global__ __launch_bounds__(256)
void gemm_wmma_f32(const float* __restrict__ A, const float* __restrict__ B,
                   float* __restrict__ C, int din, int relu_in)
{
  __shared__ float Alds[16 * (DIN + 1)];
  const int stride = din + 1;
  const int m0  = blockIdx.x * 16;
  const int tid = threadIdx.x;

  for (int i = tid; i < 16 * din; i += 256) {
    int r = i / din, d = i - r * din;
    float v = A[(size_t)(m0 + r) * din + d];
    if (relu_in) v = fmaxf(v, 0.0f);
    Alds[r * stride + d] = v;
  }
  __syncthreads();

  const int wave  = tid >> 5;
  const int lane  = tid & 31;
  const int col0  = (blockIdx.y * 8 + wave) * 16;
  const int row   = lane & 15;       // A row (M) / B,C col (N)
  const int khalf = (lane >> 4) * 2; // K split across lane halves

  v8f acc = {};
  for (int k = 0; k < din; k += 4) {
    v2f a, b;
    // A 16x4 f32 layout: element(v,lane) = A[lane%16][k + v + 2*(lane/16)]
    a.x = Alds[row * stride + k + khalf + 0];
    a.y = Alds[row * stride + k + khalf + 1];
    // B 4x16 f32 layout (symmetric): element(v,lane) = B[k + v + 2*(lane/16)][lane%16]
    b.x = B[(size_t)(k + khalf + 0) * HC + col0 + row];
    b.y = B[(size_t)(k + khalf + 1) * HC + col0 + row];
    acc = __builtin_amdgcn_wmma_f32_16x16x4_f32(
        /*neg_a=*/false, a, /*neg_b=*/false, b,
        /*c_mod=*/(short)0, acc, /*reuse_a=*/false, /*reuse_b=*/false);
  }

  // C/D 16x16 f32 layout: M = v + 8*(lane/16), N = lane%16
  const int mhalf = (lane >> 4) * 8;
#pragma unroll
  for (int v = 0; v < 8; ++v)
    C[(size_t)(m0 + v + mhalf) * HC + col0 + row] = acc[v];
}

// =====================================================================
// Per-node attention scalars: alpha_s[n,h] = sum_c xs[n,h,c]*a_s[h,c] (same a_d)
// =====================================================================
__global__ __launch_bounds__(256)
void node_alpha(const float* __restrict__ xs, const float* __restrict__ a_s,
                const float* __restrict__ a_d, float* __restrict__ alpha_s,
                float* __restrict__ alpha_d)
{
  __shared__ float ss[256], dd[256];
  const int n = blockIdx.x, j = threadIdx.x;
  float p = xs[(size_t)n * HC + j];
  ss[j] = p * a_s[j];
  dd[j] = p * a_d[j];
  __syncthreads();
  const int c = j & 63;
  for (int off = 32; off > 0; off >>= 1) {
    if (c < off) { ss[j] += ss[j + off]; dd[j] += dd[j + off]; }
    __syncthreads();
  }
  if (c == 0) {
    int h = j >> 6;
    alpha_s[n * NH + h] = ss[j];
    alpha_d[n * NH + h] = dd[j];
  }
}

// =====================================================================
// Per-relation: aerel[r,h] = sum_c (rel[r] @ We)[h,c] * a_e[h,c]   (R=64 rows)
// =====================================================================
__global__ __launch_bounds__(256)
void rel_alpha(const float* __restrict__ rel, const float* __restrict__ We,
               const float* __restrict__ a_e, float* __restrict__ aerel, int din)
{
  __shared__ float rrow[DIN];
  __shared__ float red[256];
  const int r = blockIdx.x, j = threadIdx.x;
  for (int d = j; d < din; d += 256) rrow[d] = rel[(size_t)r * din + d];
  __syncthreads();
  float p = 0.f;
  for (int d = 0; d < din; ++d) p += rrow[d] * We[(size_t)d * HC + j];
  red[j] = p * a_e[j];
  __syncthreads();
  const int c = j & 63;
  for (int off = 32; off > 0; off >>= 1) {
    if (c < off) red[j] += red[j + off];
    __syncthreads();
  }
  if (c == 0) aerel[r * NH + (j >> 6)] = red[j];
}

// =====================================================================
// Init: out = bias, lmax = 0 (ordered minimum), den = 0
// =====================================================================
__global__ void init_layer(float* __restrict__ out, const float* __restrict__ bias,
                           unsigned* __restrict__ lmax_u, float* __restrict__ den)
{
  int tid = blockIdx.x * blockDim.x + threadIdx.x;
  if (tid >= NNODES * 64) return;
  int c = tid & 63;
  out[tid] = bias[c];
  if (c < NH) {
    int n = tid >> 6;
    lmax_u[n * NH + c] = 0u;   // below the ordered encoding of every finite float
    den[n * NH + c] = 0.0f;
  }
}

__device__ __forceinline__ float edge_logit(const float* a_s, const float* a_d,
                                            const float* aerel, int s, int d,
                                            int t, int h)
{
  float l = a_s[s * NH + h] + a_d[d * NH + h] + aerel[t * NH + h];
  return (l >= 0.f) ? l : NEG_SLOPE * l;
}

// ---- pass 1: segment max over dst via monotonic-uint atomicMax ----
__global__ void edge_max(const int* __restrict__ ei, const int* __restrict__ et,
                         const float* __restrict__ a_s, const float* __restrict__ a_d,
                         const float* __restrict__ aerel, unsigned* __restrict__ lmax_u)
{
  int tid = blockIdx.x * blockDim.x + threadIdx.x;
  if (tid >= NEDGES * NH) return;
  int e = tid >> 2, h = tid & 3;
  int s = ei[e], d = ei[NEDGES + e], t = et[e];
  float l = edge_logit(a_s, a_d, aerel, s, d, t, h);
  atomicMax(&lmax_u[d * NH + h], f32_ord(l));
}

// ---- pass 2: segment sum of exp(l - lmax[dst]) ----
__global__ void edge_den(const int* __restrict__ ei, const int* __restrict__ et,
                         const float* __restrict__ a_s, const float* __restrict__ a_d,
                         const float* __restrict__ aerel,
                         const unsigned* __restrict__ lmax_u, float* __restrict__ den)
{
  int tid = blockIdx.x * blockDim.x + threadIdx.x;
  if (tid >= NEDGES * NH) return;
  int e = tid >> 2, h = tid & 3;
  int s = ei[e], d = ei[NEDGES + e], t = et[e];
  float l = edge_logit(a_s, a_d, aerel, s, d, t, h);
  float m = ord_f32(lmax_u[d * NH + h]);
  unsafeAtomicAdd(&den[d * NH + h], __expf(l - m));
}

// ---- pass 3: out[dst,c] += (1/H) * sum_h w[e,h] * xs[src,h,c]  (head-mean fused) ----
__global__ __launch_bounds__(256)
void edge_agg(const int* __restrict__ ei, const int* __restrict__ et,
              const float* __restrict__ a_s, const float* __restrict__ a_d,
              const float* __restrict__ aerel, const unsigned* __restrict__ lmax_u,
              const float* __restrict__ den, const float* __restrict__ xs,
              float* __restrict__ out)
{
  __shared__ float wsh[4][NH];
  const int le = threadIdx.x >> 6;   // 4 edges per 256-thread block
  const int c  = threadIdx.x & 63;
  const int e  = blockIdx.x * 4 + le;
  const int s = ei[e], d = ei[NEDGES + e];
  if (c < NH) {
    int t = et[e];
    float l = edge_logit(a_s, a_d, aerel, s, d, t, c);
    float m = ord_f32(lmax_u[d * NH + c]);
    wsh[le][c] = __expf(l - m) / den[d * NH + c];
  }
  __syncthreads();
  float acc = 0.f;
#pragma unroll
  for (int h = 0; h < NH; ++h)
    acc += wsh[le][h] * xs[(size_t)s * HC + h * 64 + c];
  unsafeAtomicAdd(&out[(size_t)d * 64 + c], acc * 0.25f);
}

// =====================================================================
extern "C" void kernel_launch(void* const* d_in, const int* in_sizes, int n_in,
                              void* d_out, int out_size, void* d_ws, size_t ws_size,
                              hipStream_t stream)
{
  const float* x    = (const float*)d_in[0];
  const int*   ei   = (const int*)  d_in[1];   // [2,E]: row0=src, row1=dst
  const int*   et   = (const int*)  d_in[2];
  const float* rel1 = (const float*)d_in[3];
  const float* W1   = (const float*)d_in[4];
  const float* We1  = (const float*)d_in[5];
  const float* as1  = (const float*)d_in[6];
  const float* ad1  = (const float*)d_in[7];
  const float* ae1  = (const float*)d_in[8];
  const float* b1   = (const float*)d_in[9];
  const float* rel2 = (const float*)d_in[10];
  const float* W2   = (const float*)d_in[11];
  const float* We2  = (const float*)d_in[12];
  const float* as2  = (const float*)d_in[13];
  const float* ad2  = (const float*)d_in[14];
  const float* ae2  = (const float*)d_in[15];
  const float* b2   = (const float*)d_in[16];
  float* out = (float*)d_out;

  char* ws = (char*)d_ws;
  float*    xs      = (float*)(ws);                  // 51.2 MB  [N,256]
  float*    hbuf    = (float*)(ws + 51200000);       // 12.8 MB  [N,64]
  float*    alpha_s = (float*)(ws + 64000000);       // 0.8 MB   [N,4]
  float*    alpha_d = (float*)(ws + 64800000);       // 0.8 MB
  unsigned* lmax_u  = (unsigned*)(ws + 65600000);    // 0.8 MB
  float*    den     = (float*)(ws + 66400000);       // 0.8 MB
  float*    aerel   = (float*)(ws + 67200000);       // 1 KB    [R,4]

  const dim3 gemm_grid(NNODES / 16, 2);              // 50000 % 16 == 0
  const int  eh_blocks  = (NEDGES * NH + 255) / 256;
  const int  ini_blocks = (NNODES * 64 + 255) / 256;
  const int  agg_blocks = NEDGES / 4;                // 800000 % 4 == 0

  // ---------------- Layer 1 (IN=128 -> HID=64) ----------------
  rel_alpha<<<NREL, 256, 0, stream>>>(rel1, We1, ae1, aerel, DIN);
  gemm_wmma_f32<<<gemm_grid, 256, 0, stream>>>(x, W1, xs, DIN, /*relu_in=*/0);
  node_alpha<<<NNODES, 256, 0, stream>>>(xs, as1, ad1, alpha_s, alpha_d);
  init_layer<<<ini_blocks, 256, 0, stream>>>(hbuf, b1, lmax_u, den);
  edge_max<<<eh_blocks, 256, 0, stream>>>(ei, et, alpha_s, alpha_d, aerel, lmax_u);
  edge_den<<<eh_blocks, 256, 0, stream>>>(ei, et, alpha_s, alpha_d, aerel, lmax_u, den);
  edge_agg<<<agg_blocks, 256, 0, stream>>>(ei, et, alpha_s, alpha_d, aerel, lmax_u,
                                           den, xs, hbuf);

  // ---------------- Layer 2 (HID=64 -> OUT=64), ReLU fused into GEMM load ----
  rel_alpha<<<NREL, 256, 0, stream>>>(rel2, We2, ae2, aerel, CHID);
  gemm_wmma_f32<<<gemm_grid, 256, 0, stream>>>(hbuf, W2, xs, CHID, /*relu_in=*/1);
  node_alpha<<<NNODES, 256, 0, stream>>>(xs, as2, ad2, alpha_s, alpha_d);
  init_layer<<<ini_blocks, 256, 0, stream>>>(out, b2, lmax_u, den);
  edge_max<<<eh_blocks, 256, 0, stream>>>(ei, et, alpha_s, alpha_d, aerel, lmax_u);
  edge_den<<<eh_blocks, 256, 0, stream>>>(ei, et, alpha_s, alpha_d, aerel, lmax_u, den);
  edge_agg<<<agg_blocks, 256, 0, stream>>>(ei, et, alpha_s, alpha_d, aerel, lmax_u,
                                           den, xs, out);
}